// DuctileIST_2765958939457
// MI455X (gfx1250) — compile-verified
//
#include <hip/hip_runtime.h>
#include <hip/hip_bf16.h>

// DuctileIST: per-row damage recurrence, 256 steps, output (B,256) f32.
//   lam  = 0.001 * relu(x0);  beta = 1 + 10 * relu(x1)
//   D   <- min(D + lam * (1-D)^beta, 1.0f)        (1.0f-1e-12f == 1.0f in f32)
//   out[b][t] = D after step t  (kappa == 1, no NaNs possible)
//
// Roofline: 256 MB output stream (~11us @ 23.3 TB/s) vs ~comparable
// transcendental-VALU cost -> balanced. No matrix math, so the CDNA5 feature
// that matters is the async LDS->global DMA path: stage 32x32 f32 tiles per
// wave in LDS, drain with global_store_async_from_lds_b128 so every 128B
// output cacheline is written in full, double-buffered via ASYNCcnt.
// Output is write-once streaming and larger than L2 (192MB) -> TH_STORE_NT.

#define L_STEPS   256
#define TILE_T    32           // steps per tile
#define ROWS      32           // batch rows per wave (== wave32 lanes)
#define STRIDE_F  36           // LDS row stride in floats (144B: 16B aligned, 2-way banks)
#define WAVES     4
#define BLOCK     (WAVES * 32)
#define BUF_F     (ROWS * STRIDE_F)   // 1152 floats per buffer
#define WAVE_F    (2 * BUF_F)         // double buffer per wave

__global__ __launch_bounds__(BLOCK)
void DuctileIST_kernel(const float* __restrict__ in, float* __restrict__ out)
{
    __shared__ float lds[WAVES * WAVE_F];   // 4*2304*4 = 36864 B

    const int tid  = (int)threadIdx.x;
    const int lane = tid & 31;
    const int wave = tid >> 5;
    const int b    = (int)blockIdx.x * BLOCK + tid;         // this thread's batch row
    const int b0   = (int)blockIdx.x * BLOCK + (wave << 5); // wave's first batch row

    // ---- load (B,2) input, coalesced 8B per lane -------------------------
    const float2 mo = ((const float2*)in)[b];
    const float lam  = 0.001f * fmaxf(mo.x, 0.0f);
    const float beta = 1.0f + 10.0f * fmaxf(mo.y, 0.0f);

    float D = 0.0f;

    // LDS byte-base of this wave's buffers (truncate generic ptr -> LDS offset)
    const unsigned wave_lds_base =
        (unsigned)(size_t)(&lds[wave * WAVE_F]);

    float* const wbase = &lds[wave * WAVE_F];

    const int q = lane >> 3;   // sub-row 0..3 within a 4-row store group
    const int w = lane & 7;    // 16B chunk 0..7 within a 128B row segment

    for (int tile = 0; tile < L_STEPS / TILE_T; ++tile) {
        float* const buf = wbase + (tile & 1) * BUF_F;
        const unsigned buf_base = wave_lds_base + (unsigned)((tile & 1) * BUF_F * 4);

        // Before overwriting this buffer (tile >= 2): stores of tile-2 used it.
        // Up to 8 stores from tile-1 may still be in flight; stores complete
        // in order, so asynccnt <= 8 guarantees tile-2's 8 stores are done.
        if (tile >= 2) {
            asm volatile("s_wait_asynccnt 8" ::: "memory");
        }

        // ---- compute 32 steps, stage into LDS row [lane] -----------------
        float* const row = buf + lane * STRIDE_F;
        #pragma unroll
        for (int c4 = 0; c4 < TILE_T; c4 += 4) {
            float4 v;
            #pragma unroll
            for (int k = 0; k < 4; ++k) {
                const float omd = fmaxf(1.0f - D, 1e-12f);
                // (1-D)^beta = exp2(beta * log2(1-D)) -> v_log_f32 / v_exp_f32
                const float up  = lam * __builtin_amdgcn_exp2f(
                                        beta * __builtin_amdgcn_logf(omd));
                D = fminf(D + up, 1.0f);
                ((float*)&v)[k] = D;
            }
            *(float4*)(row + c4) = v;     // ds_store_b128, 16B aligned
        }

        // LDS writes must be complete before the async DMA engine reads them.
        asm volatile("s_wait_dscnt 0" ::: "memory");

        // ---- drain tile: 8 async b128 stores, each = 4 rows x 128B -------
        // lane i -> row (4j + i/8), byte chunk (i&7)*16 : every 128B output
        // cacheline is written in full. Non-temporal: pure streaming output,
        // 256MB > 192MB L2, never re-read.
        const int t0 = tile * TILE_T;
        #pragma unroll
        for (int j = 0; j < 8; ++j) {
            const int r = 4 * j + q;
            const unsigned lds_addr =
                buf_base + (unsigned)(r * (STRIDE_F * 4) + w * 16);
            float* const gp =
                out + (size_t)(b0 + r) * L_STEPS + (size_t)(t0 + w * 4);
            asm volatile(
                "global_store_async_from_lds_b128 %0, %1, off th:TH_STORE_NT"
                :: "v"(gp), "v"(lds_addr)
                : "memory");
        }
    }

    // Drain all outstanding async stores before wave exit.
    asm volatile("s_wait_asynccnt 0" ::: "memory");
}

extern "C" void kernel_launch(void* const* d_in, const int* in_sizes, int n_in,
                              void* d_out, int out_size, void* d_ws, size_t ws_size,
                              hipStream_t stream) {
    (void)n_in; (void)d_ws; (void)ws_size; (void)out_size;
    const float* in  = (const float*)d_in[0];
    float*       out = (float*)d_out;
    const int batch  = in_sizes[0] / 2;          // (B,2) f32 -> B = 262144
    const int blocks = batch / BLOCK;            // 262144 / 128 = 2048
    DuctileIST_kernel<<<blocks, BLOCK, 0, stream>>>(in, out);
}